// CorrLSTM_upsample_resize_29609504538954
// MI455X (gfx1250) — compile-verified
//
#include <hip/hip_runtime.h>
#include <math.h>

typedef __attribute__((ext_vector_type(2))) float v2f;
typedef __attribute__((ext_vector_type(8))) float v8f;

#define P    7
#define P2   49
#define OC   196     // 4*P2 conv output channels
#define ICV  98      // 2*P2 conv input channels
#define KPAD 100     // padded input-channel count (mult of 4)
#define MT   13      // ceil(OC/16)
#define MPAD 208     // MT*16
#define WPROW 900    // 9*KPAD floats per padded output-channel row
#define NR   4       // output rows per workgroup (N-tiles per A fragment)

// ---------------------------------------------------------------- utilities
__global__ void zero_f32(float* __restrict__ p, int n) {
  int i = blockIdx.x * blockDim.x + threadIdx.x;
  if (i < n) p[i] = 0.f;
}

// Repack OIHW weights (196,98,3,3) -> Wp[o][pos][i], o padded to 208, i to 100.
__global__ void repack_w(const float* __restrict__ w, float* __restrict__ wp) {
  int idx = blockIdx.x * blockDim.x + threadIdx.x;
  if (idx >= MPAD * WPROW) return;
  int o   = idx / WPROW;
  int r   = idx % WPROW;
  int pos = r / KPAD;        // ky*3+kx
  int i   = r % KPAD;
  float v = 0.f;
  if (o < OC && i < ICV) v = w[(o * ICV + i) * 9 + pos];
  wp[idx] = v;
}

// ------------------------------------------------- collapsed correlation
// Upsample(s) -> 7x7 correlate -> downsample(s) collapses to base-resolution
// shifts floor((p-3)/s): ND=4,HLO=2 for s=2 (shifts -2..1); ND=2,HLO=1 for s=4.
template <int ND, int HLO>
__global__ __launch_bounds__(256) void corr_kernel(
    const float* __restrict__ f1, const float* __restrict__ f2,
    float* __restrict__ corr, int C, int H, int W, int cPerWG, int nsplit) {
  constexpr int ROWS = 16 + ND - 1;
  constexpr int ND2  = ND * ND;
  constexpr int CC   = 8;
  __shared__ float sF2[CC * ROWS * ROWS];
  const int tid = threadIdx.x;
  const int tx = tid & 15, ty = tid >> 4;
  const int bx = blockIdx.x, by = blockIdx.y;
  const int b     = blockIdx.z / nsplit;
  const int split = blockIdx.z % nsplit;
  const int gx = bx * 16 + tx, gy = by * 16 + ty;
  const int HW = H * W;

  float acc[ND2];
#pragma unroll
  for (int d = 0; d < ND2; ++d) acc[d] = 0.f;

  const int cbeg = split * cPerWG;
  for (int c0 = cbeg; c0 < cbeg + cPerWG; c0 += CC) {
    for (int e = tid; e < CC * ROWS * ROWS; e += 256) {
      int cc = e / (ROWS * ROWS);
      int r  = e % (ROWS * ROWS);
      int ry = r / ROWS, rx = r % ROWS;
      int sy = by * 16 + ry - HLO;
      int sx = bx * 16 + rx - HLO;
      float v = 0.f;
      if (sy >= 0 && sy < H && sx >= 0 && sx < W)
        v = f2[(size_t)(b * C + c0 + cc) * HW + sy * W + sx];
      sF2[e] = v;
    }
    __syncthreads();
#pragma unroll
    for (int cc = 0; cc < CC; ++cc) {
      float f1v = f1[(size_t)(b * C + c0 + cc) * HW + gy * W + gx];
      const float* s = &sF2[cc * ROWS * ROWS];
#pragma unroll
      for (int dy = 0; dy < ND; ++dy)
#pragma unroll
        for (int dx = 0; dx < ND; ++dx)
          acc[dy * ND + dx] =
              fmaf(f1v, s[(ty + dy) * ROWS + (tx + dx)], acc[dy * ND + dx]);
    }
    __syncthreads();
  }
#pragma unroll
  for (int d = 0; d < ND2; ++d)
    atomicAdd(&corr[(size_t)(b * ND2 + d) * HW + gy * W + gx], acc[d]);
}

// ------------------------------------------------- conv3x3 (WMMA) + LSTM cell
__device__ __forceinline__ float sigmoidf_(float x) {
  return 1.f / (1.f + __expf(-x));
}

template <int ND, int LOG2S>
__global__ __launch_bounds__(256) void conv_lstm(
    const float* __restrict__ corr, const float* __restrict__ hpre,
    const float* __restrict__ cpre, const float* __restrict__ wp,
    const float* __restrict__ bias, float* __restrict__ outh,
    float* __restrict__ outc, int H, int W) {
  constexpr int ND2 = ND * ND;
  __shared__ float sIn[KPAD * (NR + 2) * 18];  // [i][row y0-1..y0+4][x0-1..x0+16]
  __shared__ float sConv[MPAD * (NR * 16)];    // conv output (+bias), 208 x 64
  const int tid = threadIdx.x;
  const int x0 = blockIdx.x * 16;
  const int y0 = blockIdx.y * NR;
  const int b  = blockIdx.z;
  const int HW = H * W;

  // Stage im2col source: 49 corr channels (via p->shift map, leaky ReLU),
  // then 49 h channels, then 2 zero-pad channels. Borders zero-padded.
  for (int e = tid; e < KPAD * (NR + 2) * 18; e += 256) {
    int i   = e / ((NR + 2) * 18);
    int r   = e % ((NR + 2) * 18);
    int ry  = r / 18;
    int xp  = r % 18;
    int gy  = y0 + ry - 1;
    int gx  = x0 + xp - 1;
    float v = 0.f;
    if (gy >= 0 && gy < H && gx >= 0 && gx < W && i < ICV) {
      if (i < P2) {
        int pi = i / P, pj = i % P;
        int d  = ((pi + 1) >> LOG2S) * ND + ((pj + 1) >> LOG2S);
        float c = corr[(size_t)(b * ND2 + d) * HW + gy * W + gx];
        v = (c >= 0.f) ? c : 0.01f * c;
      } else {
        v = hpre[(size_t)(b * P2 + (i - P2)) * HW + gy * W + gx];
      }
    }
    sIn[e] = v;
  }
  __syncthreads();

  // GEMM: D[196 x 64] = W[196 x 882] * im2col[882 x 64] via f32 WMMA (K=4).
  // One A fragment feeds NR=4 WMMAs (one per output row) -> 4x A reuse.
  const int wave  = tid >> 5;
  const int lane  = tid & 31;
  const int mrow  = lane & 15;   // A row within tile
  const int khalf = lane >> 4;   // 0: supplies K+0/K+1, 1: K+2/K+3
  const int n     = lane & 15;   // B/D column

  for (int t = wave; t < MT; t += 8) {
    const int m0 = t * 16;
    v8f acc[NR];
#pragma unroll
    for (int r = 0; r < NR; ++r)
      acc[r] = (v8f){0.f, 0.f, 0.f, 0.f, 0.f, 0.f, 0.f, 0.f};
    const float* wb = wp + (size_t)(m0 + mrow) * WPROW + khalf * 2;
#pragma unroll 1
    for (int pos = 0; pos < 9; ++pos) {
      const int ky = pos / 3, kx = pos % 3;
      const float* wpos = wb + pos * KPAD;
      const float* spos = &sIn[ky * 18 + n + kx];
#pragma unroll
      for (int i0 = 0; i0 < KPAD; i0 += 4) {
        v2f a = *(const v2f*)(wpos + i0);       // A[m][k], k = i0+2*khalf(+1)
        int kb = i0 + khalf * 2;
        const float* sp0 = spos + kb * ((NR + 2) * 18);
        const float* sp1 = spos + (kb + 1) * ((NR + 2) * 18);
#pragma unroll
        for (int r = 0; r < NR; ++r) {
          v2f bf;
          bf.x = sp0[r * 18];                   // B[k][row r, col n]
          bf.y = sp1[r * 18];
          acc[r] = __builtin_amdgcn_wmma_f32_16x16x4_f32(
              false, a, false, bf, (short)0, acc[r], false, false);
        }
      }
    }
#pragma unroll
    for (int rr = 0; rr < 8; ++rr) {
      int m = m0 + rr + (khalf << 3);
      float bv = (m < OC) ? bias[m] : 0.f;
#pragma unroll
      for (int r = 0; r < NR; ++r)
        sConv[m * (NR * 16) + r * 16 + n] = acc[r][rr] + bv;
    }
  }
  __syncthreads();

  // LSTM gate math for the 64 pixels of this patch.
  const int q  = tid & 63;            // pixel: row q>>4, col q&15
  const int gy = y0 + (q >> 4);
  const int gx = x0 + (q & 15);
  for (int ch = tid >> 6; ch < P2; ch += 4) {
    float ci = sConv[ch * 64 + q];
    float cf = sConv[(P2 + ch) * 64 + q];
    float co = sConv[(2 * P2 + ch) * 64 + q];
    float cg = sConv[(3 * P2 + ch) * 64 + q];
    float ig = sigmoidf_(ci);
    float fg = sigmoidf_(cf);
    float og = sigmoidf_(co);
    float g  = tanhf(cg);
    size_t idx = (size_t)(b * P2 + ch) * HW + gy * W + gx;
    float cn = fg * cpre[idx] + ig * g;
    float hn = og * tanhf(cn);
    outc[idx] = cn;
    outh[idx] = hn;
  }
}

// ---------------------------------------------------------------- launcher
extern "C" void kernel_launch(void* const* d_in, const int* in_sizes, int n_in,
                              void* d_out, int out_size, void* d_ws,
                              size_t ws_size, hipStream_t stream) {
  const float* x0  = (const float*)d_in[0];   // (2,512,64,64)
  const float* x1  = (const float*)d_in[1];   // (2,1024,32,32)
  const float* xp0 = (const float*)d_in[2];
  const float* xp1 = (const float*)d_in[3];
  const float* h0  = (const float*)d_in[4];   // (2,49,64,64)
  const float* c0  = (const float*)d_in[5];
  const float* h1  = (const float*)d_in[6];   // (2,49,32,32)
  const float* c1  = (const float*)d_in[7];
  const float* w0  = (const float*)d_in[8];   // (196,98,3,3)
  const float* b0  = (const float*)d_in[9];
  const float* w1  = (const float*)d_in[10];
  const float* b1  = (const float*)d_in[11];

  float* ws    = (float*)d_ws;
  float* corr0 = ws;                          // 2*16*64*64 = 131072
  float* corr1 = corr0 + 131072;              // 2*4*32*32  = 8192
  float* wp0   = corr1 + 8192;                // 208*900    = 187200
  float* wp1   = wp0 + 187200;

  float* out = (float*)d_out;
  float* h0n = out;                           // 2*49*64*64 = 401408
  float* c0n = out + 401408;
  float* h1n = out + 802816;                  // 2*49*32*32 = 100352
  float* c1n = out + 903168;

  int zn = 131072 + 8192;
  zero_f32<<<(zn + 255) / 256, 256, 0, stream>>>(ws, zn);
  repack_w<<<(MPAD * WPROW + 255) / 256, 256, 0, stream>>>(w0, wp0);
  repack_w<<<(MPAD * WPROW + 255) / 256, 256, 0, stream>>>(w1, wp1);

  // level 0: scale 2 -> shifts {-2,-1,0,1}; level 1: scale 4 -> {-1,0}
  corr_kernel<4, 2><<<dim3(4, 4, 2 * 8), 256, 0, stream>>>(
      x0, xp0, corr0, 512, 64, 64, 64, 8);
  corr_kernel<2, 1><<<dim3(2, 2, 2 * 8), 256, 0, stream>>>(
      x1, xp1, corr1, 1024, 32, 32, 128, 8);

  conv_lstm<4, 1><<<dim3(4, 64 / NR, 2), 256, 0, stream>>>(
      corr0, h0, c0, wp0, b0, h0n, c0n, 64, 64);
  conv_lstm<2, 2><<<dim3(2, 32 / NR, 2), 256, 0, stream>>>(
      corr1, h1, c1, wp1, b1, h1n, c1n, 32, 32);
}